// Mamba_47450798686480
// MI455X (gfx1250) — compile-verified
//
#include <hip/hip_runtime.h>
#include <hip/hip_bf16.h>
#include <math.h>

// Problem dimensions (fixed by the reference).
#define L_      2048
#define DMODEL  768
#define NLAYER  2
#define DINNER  1536
#define DSTATE  16
#define DTRANK  48
#define DCONV   4
#define OUTDIM  128
#define PROJW   (DTRANK + 2 * DSTATE)   // 80

typedef float v2f __attribute__((ext_vector_type(2)));
typedef float v8f __attribute__((ext_vector_type(8)));
typedef unsigned int v4u __attribute__((ext_vector_type(4)));
typedef int v4i __attribute__((ext_vector_type(4)));
typedef int v8i __attribute__((ext_vector_type(8)));

// ---------------------------------------------------------------------------
// TDM: async 2D tile load Global -> LDS (tensor_load_to_lds, TENSORcnt).
// D# layout per CDNA5 ISA 8.3/8.4.  2-D tensor: groups 2/3 zeroed.
// lds_addr      : LDS byte offset of destination
// gaddr         : global address of the TILE start (not tensor start)
// tile_w/tile_h : tile dims in elements (dim0 = contiguous)
// tensor_w/h    : tensor dims in elements (OOB clamp)
// row_stride    : elements between consecutive rows
// ---------------------------------------------------------------------------
__device__ static inline void tdm_load_2d_f32(unsigned lds_addr, const void* gaddr,
                                              unsigned tile_w, unsigned tile_h,
                                              unsigned tensor_w, unsigned tensor_h,
                                              unsigned row_stride)
{
    unsigned long long ga = (unsigned long long)(size_t)gaddr;
    v4u g0;
    g0[0] = 1u;                                          // count=1 (valid), user D#
    g0[1] = lds_addr;                                    // [63:32] lds_addr
    g0[2] = (unsigned)(ga & 0xFFFFFFFFu);                // global_addr[31:0]
    g0[3] = ((unsigned)(ga >> 32) & 0x01FFFFFFu)         // global_addr[56:32]
            | (2u << 30);                                // type = 2 ("image")
    v8i g1;
    g1[0] = (int)(2u << 16);                             // data_size = 2 -> 4 bytes
    g1[1] = (int)((tensor_w & 0xFFFFu) << 16);           // tensor_dim0[15:0]
    g1[2] = (int)(((tensor_w >> 16) & 0xFFFFu)
            | ((tensor_h & 0xFFFFu) << 16));             // tensor_dim0[31:16]|dim1[15:0]
    g1[3] = (int)(((tensor_h >> 16) & 0xFFFFu)
            | ((tile_w & 0xFFFFu) << 16));               // dim1[31:16]|tile_dim0
    g1[4] = (int)(tile_h & 0xFFFFu);                     // tile_dim1 (tile_dim2=0)
    g1[5] = (int)row_stride;                             // tensor_dim0_stride[31:0]
    g1[6] = 0;                                           // stride[47:32]|dim1_stride lo
    g1[7] = 0;
    v4i z4 = {0, 0, 0, 0};
#if __clang_major__ >= 23
    v8i z8 = {0, 0, 0, 0, 0, 0, 0, 0};
    __builtin_amdgcn_tensor_load_to_lds(g0, g1, z4, z4, z8, 0);
#else
    __builtin_amdgcn_tensor_load_to_lds(g0, g1, z4, z4, 0);
#endif
}

// ---------------------------------------------------------------------------
// FP32 WMMA GEMM, 16x16 tile per wave (used only for N % 64 != 0, i.e. x_proj).
// ---------------------------------------------------------------------------
__global__ __launch_bounds__(256) void gemm_wmma_f32_16(
    const float* __restrict__ A, const float* __restrict__ W,
    float* __restrict__ C, int M, int N, int K,
    int lda, int ldw, int ldc, int accumulate)
{
    const int wave   = (int)(blockIdx.x * 8 + (threadIdx.x >> 5));
    const int lane   = (int)(threadIdx.x & 31);
    const int tilesN = N >> 4;
    const int tM     = (wave / tilesN) << 4;
    const int tN     = (wave % tilesN) << 4;
    if (tM >= M) return;                      // whole wave exits together

    const int lm = lane & 15;
    const int kh = lane >> 4;                 // K half: 0 -> K{0,1}, 1 -> K{2,3}
    const int mb = kh << 3;

    const float* __restrict__ Arow = A + (size_t)(tM + lm) * lda + 2 * kh;
    const float* __restrict__ Wrow = W + (size_t)(tN + lm) * ldw + 2 * kh;

    v8f c = {};
    if (accumulate) {
#pragma unroll
        for (int r = 0; r < 8; ++r)
            c[r] = C[(size_t)(tM + mb + r) * ldc + tN + lm];
    }
#pragma unroll 4
    for (int k = 0; k < K; k += 4) {
        if ((k & 31) == 0) {
            __builtin_prefetch(Arow + k + 32, 0, 1);
            __builtin_prefetch(Wrow + k + 32, 0, 1);
        }
        v2f a, b;
        a.x = Arow[k];  a.y = Arow[k + 1];
        b.x = Wrow[k];  b.y = Wrow[k + 1];
        c = __builtin_amdgcn_wmma_f32_16x16x4_f32(false, a, false, b,
                                                  (short)0, c, false, false);
    }
#pragma unroll
    for (int r = 0; r < 8; ++r)
        C[(size_t)(tM + mb + r) * ldc + tN + lm] = c[r];
}

// ---------------------------------------------------------------------------
// FP32 WMMA GEMM, register-blocked: 32(M) x 64(N) tile per wave.
// 8 accumulators; per k-step: 2 A-fragments + 4 B-fragments -> 8 WMMAs
// (6 b64 loads per 8 matrix ops).  Requires M%32==0 and N%64==0.
// ---------------------------------------------------------------------------
__global__ __launch_bounds__(256) void gemm_wmma_f32_blk(
    const float* __restrict__ A, const float* __restrict__ W,
    float* __restrict__ C, int M, int N, int K,
    int lda, int ldw, int ldc, int accumulate)
{
    const int wave   = (int)(blockIdx.x * 8 + (threadIdx.x >> 5));
    const int lane   = (int)(threadIdx.x & 31);
    const int tilesN = N >> 6;                // 64-wide tiles
    const int tM     = (wave / tilesN) << 5;  // 32-tall tiles
    const int tN     = (wave % tilesN) << 6;
    if (tM >= M) return;

    const int lm = lane & 15;
    const int kh = lane >> 4;
    const int mb = kh << 3;

    const float* __restrict__ A0 = A + (size_t)(tM + lm) * lda + 2 * kh;
    const float* __restrict__ A1 = A0 + (size_t)16 * lda;
    const float* __restrict__ W0 = W + (size_t)(tN + lm) * ldw + 2 * kh;

    v8f c[2][4];
#pragma unroll
    for (int mi = 0; mi < 2; ++mi)
#pragma unroll
        for (int j = 0; j < 4; ++j) {
            c[mi][j] = (v8f){};
            if (accumulate) {
#pragma unroll
                for (int r = 0; r < 8; ++r)
                    c[mi][j][r] =
                        C[(size_t)(tM + mi * 16 + mb + r) * ldc + tN + j * 16 + lm];
            }
        }

#pragma unroll 2
    for (int k = 0; k < K; k += 4) {
        if ((k & 31) == 0) {
            __builtin_prefetch(A0 + k + 32, 0, 1);
            __builtin_prefetch(W0 + k + 32, 0, 1);
        }
        v2f a0, a1;
        a0.x = A0[k]; a0.y = A0[k + 1];
        a1.x = A1[k]; a1.y = A1[k + 1];
#pragma unroll
        for (int j = 0; j < 4; ++j) {
            v2f b;
            b.x = W0[(size_t)j * 16 * ldw + k];
            b.y = W0[(size_t)j * 16 * ldw + k + 1];
            c[0][j] = __builtin_amdgcn_wmma_f32_16x16x4_f32(false, a0, false, b,
                                                            (short)0, c[0][j],
                                                            false, false);
            c[1][j] = __builtin_amdgcn_wmma_f32_16x16x4_f32(false, a1, false, b,
                                                            (short)0, c[1][j],
                                                            false, false);
        }
    }

#pragma unroll
    for (int mi = 0; mi < 2; ++mi)
#pragma unroll
        for (int j = 0; j < 4; ++j)
#pragma unroll
            for (int r = 0; r < 8; ++r)
                C[(size_t)(tM + mi * 16 + mb + r) * ldc + tN + j * 16 + lm] =
                    c[mi][j][r];
}

// ---------------------------------------------------------------------------
// RMSNorm: one 256-thread block per row of D elements.
// ---------------------------------------------------------------------------
__global__ __launch_bounds__(256) void rmsnorm_kernel(
    const float* __restrict__ x, const float* __restrict__ w,
    float* __restrict__ o, int D)
{
    __shared__ float red[256];
    const int row = blockIdx.x;
    const float* xr = x + (size_t)row * D;
    float acc = 0.f;
    for (int i = threadIdx.x; i < D; i += 256) { float v = xr[i]; acc += v * v; }
    red[threadIdx.x] = acc;
    __syncthreads();
    for (int s = 128; s > 0; s >>= 1) {
        if ((int)threadIdx.x < s) red[threadIdx.x] += red[threadIdx.x + s];
        __syncthreads();
    }
    const float scale = rsqrtf(red[0] / (float)D + 1e-5f);
    for (int i = threadIdx.x; i < D; i += 256)
        o[(size_t)row * D + i] = xr[i] * scale * w[i];
}

// ---------------------------------------------------------------------------
// Causal depthwise conv1d (kernel=4) + bias + SiLU.
// ---------------------------------------------------------------------------
__global__ __launch_bounds__(256) void conv_silu_kernel(
    const float* __restrict__ xz, const float* __restrict__ cw,
    const float* __restrict__ cb, float* __restrict__ xc)
{
    const int idx = (int)(blockIdx.x * 256 + threadIdx.x);
    if (idx >= L_ * DINNER) return;
    const int c = idx % DINNER;
    const int t = idx / DINNER;
    float acc = cb[c];
#pragma unroll
    for (int j = 0; j < DCONV; ++j) {
        const int tt = t - (DCONV - 1) + j;
        if (tt >= 0) acc += cw[c * DCONV + j] * xz[(size_t)tt * (2 * DINNER) + c];
    }
    xc[idx] = acc / (1.f + __expf(-acc));     // SiLU
}

// delta = softplus(delta_raw + dt_bias)
__global__ __launch_bounds__(256) void bias_softplus_kernel(
    float* __restrict__ delta, const float* __restrict__ dtb)
{
    const int idx = (int)(blockIdx.x * 256 + threadIdx.x);
    if (idx >= L_ * DINNER) return;
    const float v = delta[idx] + dtb[idx % DINNER];
    delta[idx] = (v > 20.f) ? v : log1pf(__expf(v));
}

// ---------------------------------------------------------------------------
// Selective scan.  One thread per channel d (16 states in registers), marching
// t = 0..L-1.  The shared B/C rows for a TCHUNK-step window are DMA'd into LDS
// by the Tensor Data Mover (64 x 32 f32 tile out of proj, row stride 80),
// completed with s_wait_tensorcnt + block barrier.
// ---------------------------------------------------------------------------
#define TCHUNK 64
__global__ __launch_bounds__(256) void scan_kernel(
    const float* __restrict__ delta, const float* __restrict__ xc,
    const float* __restrict__ proj,  const float* __restrict__ a_log,
    const float* __restrict__ Dv,    float* __restrict__ y)
{
    __shared__ float sBC[TCHUNK * 2 * DSTATE];   // [t][0:16]=B, [t][16:32]=C
    const int d = (int)(blockIdx.x * 256 + threadIdx.x);   // 6 blocks * 256 = 1536

    float Ad[DSTATE], s[DSTATE];
#pragma unroll
    for (int n = 0; n < DSTATE; ++n) {
        Ad[n] = -__expf(a_log[(size_t)d * DSTATE + n]);    // A = -exp(A_log)
        s[n]  = 0.f;
    }
    const float Dd = Dv[d];
    // Flat LDS-aperture pointers keep the LDS byte offset in addr[31:0].
    const unsigned lds_off = (unsigned)(size_t)&sBC[0];

    for (int t0 = 0; t0 < L_; t0 += TCHUNK) {
        __syncthreads();                       // previous window fully consumed
        if (threadIdx.x < 32) {                // wave 0 drives the TDM
            tdm_load_2d_f32(lds_off,
                            proj + (size_t)t0 * PROJW + DTRANK,
                            /*tile_w=*/2 * DSTATE, /*tile_h=*/TCHUNK,
                            /*tensor_w=*/2 * DSTATE,
                            /*tensor_h=*/(unsigned)(L_ - t0),
                            /*row_stride=*/PROJW);
            __builtin_amdgcn_s_wait_tensorcnt(0);
        }
        __syncthreads();                       // publish LDS tile to all waves

        for (int tl = 0; tl < TCHUNK; ++tl) {
            const int t = t0 + tl;
            const float dt = delta[(size_t)t * DINNER + d];
            const float u  = xc[(size_t)t * DINNER + d];
            const float du = dt * u;
            float yv = 0.f;
#pragma unroll
            for (int n = 0; n < DSTATE; ++n) {
                const float dA = __expf(dt * Ad[n]);
                s[n] = dA * s[n] + du * sBC[tl * 32 + n];
                yv  += s[n] * sBC[tl * 32 + DSTATE + n];
            }
            y[(size_t)t * DINNER + d] = yv + u * Dd;
        }
    }
}

// y *= silu(res), res = second half of the in_proj output
__global__ __launch_bounds__(256) void gate_kernel(
    float* __restrict__ y, const float* __restrict__ xz)
{
    const int idx = (int)(blockIdx.x * 256 + threadIdx.x);
    if (idx >= L_ * DINNER) return;
    const int c = idx % DINNER;
    const int t = idx / DINNER;
    const float r = xz[(size_t)t * (2 * DINNER) + DINNER + c];
    y[idx] *= r / (1.f + __expf(-r));
}

// ---------------------------------------------------------------------------
static inline void launch_gemm(const float* A, const float* W, float* C,
                               int M, int N, int K, int lda, int ldw, int ldc,
                               int accumulate, hipStream_t stream)
{
    if ((N & 63) == 0 && (M & 31) == 0) {
        const int tiles  = (M / 32) * (N / 64);
        const int blocks = (tiles + 7) / 8;
        gemm_wmma_f32_blk<<<blocks, 256, 0, stream>>>(A, W, C, M, N, K,
                                                      lda, ldw, ldc, accumulate);
    } else {
        const int tiles  = (M / 16) * (N / 16);
        const int blocks = (tiles + 7) / 8;
        gemm_wmma_f32_16<<<blocks, 256, 0, stream>>>(A, W, C, M, N, K,
                                                     lda, ldw, ldc, accumulate);
    }
}

extern "C" void kernel_launch(void* const* d_in, const int* in_sizes, int n_in,
                              void* d_out, int out_size, void* d_ws, size_t ws_size,
                              hipStream_t stream)
{
    const float* x    = (const float*)d_in[0];
    const float* in_w = (const float*)d_in[1];
    const float* cw   = (const float*)d_in[2];
    const float* cb   = (const float*)d_in[3];
    const float* xpw  = (const float*)d_in[4];
    const float* dtw  = (const float*)d_in[5];
    const float* dtb  = (const float*)d_in[6];
    const float* alog = (const float*)d_in[7];
    const float* Dvp  = (const float*)d_in[8];
    const float* ow   = (const float*)d_in[9];
    const float* nw   = (const float*)d_in[10];
    const float* nfw  = (const float*)d_in[11];
    const float* hw   = (const float*)d_in[12];
    float* out = (float*)d_out;

    // Workspace layout (floats)
    float* ws    = (float*)d_ws;
    float* x_cur = ws;                                     // L*DMODEL
    float* xn    = x_cur + (size_t)L_ * DMODEL;            // L*DMODEL
    float* xz    = xn    + (size_t)L_ * DMODEL;            // L*2*DINNER
    float* xc    = xz    + (size_t)L_ * 2 * DINNER;        // L*DINNER
    float* proj  = xc    + (size_t)L_ * DINNER;            // L*PROJW
    float* delta = proj  + (size_t)L_ * PROJW;             // L*DINNER
    float* yb    = delta + (size_t)L_ * DINNER;            // L*DINNER

    hipMemcpyAsync(x_cur, x, sizeof(float) * (size_t)L_ * DMODEL,
                   hipMemcpyDeviceToDevice, stream);

    const int nEW = (L_ * DINNER + 255) / 256;             // elementwise grid

    for (int layer = 0; layer < NLAYER; ++layer) {
        const float* w_in  = in_w + (size_t)layer * 2 * DINNER * DMODEL;
        const float* w_cv  = cw   + (size_t)layer * DINNER * DCONV;
        const float* b_cv  = cb   + (size_t)layer * DINNER;
        const float* w_xp  = xpw  + (size_t)layer * PROJW * DINNER;
        const float* w_dt  = dtw  + (size_t)layer * DINNER * DTRANK;
        const float* b_dt  = dtb  + (size_t)layer * DINNER;
        const float* a_l   = alog + (size_t)layer * DINNER * DSTATE;
        const float* d_l   = Dvp  + (size_t)layer * DINNER;
        const float* w_out = ow   + (size_t)layer * DMODEL * DINNER;
        const float* w_n   = nw   + (size_t)layer * DMODEL;

        // pre-norm
        rmsnorm_kernel<<<L_, 256, 0, stream>>>(x_cur, w_n, xn, DMODEL);
        // in_proj: (L,768) x (3072,768)^T -> (L,3072)        [blocked WMMA]
        launch_gemm(xn, w_in, xz, L_, 2 * DINNER, DMODEL,
                    DMODEL, DMODEL, 2 * DINNER, 0, stream);
        // causal depthwise conv + SiLU
        conv_silu_kernel<<<nEW, 256, 0, stream>>>(xz, w_cv, b_cv, xc);
        // x_proj: (L,1536) x (80,1536)^T -> (L,80)           [16x16 WMMA]
        launch_gemm(xc, w_xp, proj, L_, PROJW, DINNER,
                    DINNER, DINNER, PROJW, 0, stream);
        // dt_proj: (L,48) x (1536,48)^T -> (L,1536)          [blocked WMMA]
        launch_gemm(proj, w_dt, delta, L_, DINNER, DTRANK,
                    PROJW, DTRANK, DINNER, 0, stream);
        bias_softplus_kernel<<<nEW, 256, 0, stream>>>(delta, b_dt);
        // selective scan (TDM-staged B/C tiles)
        scan_kernel<<<DINNER / 256, 256, 0, stream>>>(delta, xc, proj,
                                                      a_l, d_l, yb);
        gate_kernel<<<nEW, 256, 0, stream>>>(yb, xz);
        // out_proj with fused residual: x_cur += (L,1536) x (768,1536)^T
        launch_gemm(yb, w_out, x_cur, L_, DMODEL, DINNER,
                    DINNER, DINNER, DMODEL, 1, stream);
    }

    // final norm + lm head: (L,768) x (128,768)^T -> (L,128)
    rmsnorm_kernel<<<L_, 256, 0, stream>>>(x_cur, nfw, xn, DMODEL);
    launch_gemm(xn, hw, out, L_, OUTDIM, DMODEL,
                DMODEL, DMODEL, OUTDIM, 0, stream);
}